// Cross_Entropy_s9_55525337202956
// MI455X (gfx1250) — compile-verified
//
#include <hip/hip_runtime.h>

// Clang ext-vector types (WMMA operands must be clang vectors, and
// __builtin_nontemporal_load requires scalar/vector pointee).
typedef __attribute__((ext_vector_type(2))) float v2f;
typedef __attribute__((ext_vector_type(4))) float v4f;
typedef __attribute__((ext_vector_type(8))) float v8f;

#define FP_ALPHA 0.25f
#define FP_CORR  1.0f

// Per-row focal loss, 2-class closed form:
//   d = p1 - p0 ; prb1 = sigmoid(d) ; prb0 = 1 - prb1
//   lp1 = log(prb1) ; lp0 = log(prb0)
//   focal0 = (1-prb0)^2 = prb1^2 ; focal1 = prb0^2   (GAMMA == 2)
//   loss = -(oh0*lp0*prb1^2 + oh1*lp1*prb0^2)
__device__ __forceinline__ float focal_row(float p0, float p1, float g) {
    const float d   = p1 - p0;
    const float t   = (g >= 0.5f) ? 1.0f : 0.0f;
    const float oh1 = t * FP_ALPHA;
    const float oh0 = (1.0f - oh1) * (1.0f - FP_ALPHA);
    const float e   = __expf(-d);           // v_exp_f32
    const float sig = 1.0f / (1.0f + e);    // prb1
    const float q   = 1.0f - sig;           // prb0
    const float lp1 = __logf(sig);          // v_log_f32
    const float lp0 = __logf(q);
    return -(oh0 * lp0 * sig * sig + oh1 * lp1 * q * q);
}

// Kernel 1: grid-stride streaming pass with b128 NT loads; wave32 reduction
// via V_WMMA_F32_16X16X4_F32 (A = lane partials, B = ones); per-block partial
// sum written to workspace (deterministic — no float atomics).
template <int BLOCK>
__global__ __launch_bounds__(BLOCK) void focal_partial_kernel(
    const float* __restrict__ pred, const float* __restrict__ gold,
    float* __restrict__ partial, int n) {
    const int tid    = blockIdx.x * BLOCK + threadIdx.x;
    const int stride = gridDim.x * BLOCK;
    const int groups = n >> 2;  // 4 rows per iteration

    const v4f* __restrict__ pred4 = reinterpret_cast<const v4f*>(pred);
    const v4f* __restrict__ gold4 = reinterpret_cast<const v4f*>(gold);

    float acc = 0.0f;
    for (int g = tid; g < groups; g += stride) {
        // rows 4g..4g+3: 2x global_load_b128 (pred) + 1x global_load_b128 (gold), NT hint
        const v4f pa = __builtin_nontemporal_load(&pred4[2 * g + 0]);
        const v4f pb = __builtin_nontemporal_load(&pred4[2 * g + 1]);
        const v4f gg = __builtin_nontemporal_load(&gold4[g]);
        acc += focal_row(pa.x, pa.y, gg.x);
        acc += focal_row(pa.z, pa.w, gg.y);
        acc += focal_row(pb.x, pb.y, gg.z);
        acc += focal_row(pb.z, pb.w, gg.w);
    }
    // Tail (n not a multiple of 4) — empty for the reference shape.
    for (int r = (groups << 2) + tid; r < n; r += stride) {
        acc += focal_row(pred[2 * r], pred[2 * r + 1], gold[r]);
    }

    // ---- Wave reduction with one WMMA (EXEC is all-1s here: control flow
    // above is reconverged and trip counts are per-thread uniform).
    // A (16x4 f32, 2 VGPRs/lane): lane L<16 holds A[L][0],A[L][1];
    // lane L>=16 holds A[L-16][2],A[L-16][3]. Put partial in slot 0, zero in slot 1.
    // B = ones(4x16)  =>  D[m][n] = acc_m + acc_{m+16} for every n.
    v2f a; a.x = acc;  a.y = 0.0f;
    v2f b; b.x = 1.0f; b.y = 1.0f;
    v8f c = {};
    v8f dm = __builtin_amdgcn_wmma_f32_16x16x4_f32(
        /*neg_a=*/false, a, /*neg_b=*/false, b,
        /*c_mod=*/(short)0, c, /*reuse_a=*/false, /*reuse_b=*/false);

    // D layout: lanes 0-15 VGPR r = D[r][lane]; lanes 16-31 VGPR r = D[r+8][lane-16].
    // Sum the 8 D regs: lane0 -> sum of rows 0..7, lane16 -> rows 8..15.
    float s = dm[0] + dm[1] + dm[2] + dm[3] + dm[4] + dm[5] + dm[6] + dm[7];
    s += __shfl_xor(s, 16, 32);  // combine the two half-wave row groups

    __shared__ float lds[BLOCK / 32];
    const int lane = threadIdx.x & 31;
    const int wave = threadIdx.x >> 5;
    if (lane == 0) lds[wave] = s;
    __syncthreads();
    if (threadIdx.x == 0) {
        float bs = 0.0f;
#pragma unroll
        for (int i = 0; i < BLOCK / 32; ++i) bs += lds[i];
        partial[blockIdx.x] = bs;
    }
}

// Kernel 2: deterministic fixed-order tree reduction of the block partials.
__global__ __launch_bounds__(256) void focal_final_kernel(
    const float* __restrict__ partial, float* __restrict__ out, int nblocks) {
    __shared__ float lds[256];
    float s = 0.0f;
    for (int i = threadIdx.x; i < nblocks; i += 256) s += partial[i];
    lds[threadIdx.x] = s;
    __syncthreads();
#pragma unroll
    for (int off = 128; off > 0; off >>= 1) {
        if (threadIdx.x < (unsigned)off) lds[threadIdx.x] += lds[threadIdx.x + off];
        __syncthreads();
    }
    if (threadIdx.x == 0) out[0] = lds[0] * FP_CORR;
}

extern "C" void kernel_launch(void* const* d_in, const int* in_sizes, int n_in,
                              void* d_out, int out_size, void* d_ws, size_t ws_size,
                              hipStream_t stream) {
    const float* pred = (const float*)d_in[0];  // [N, 2] f32
    const float* gold = (const float*)d_in[1];  // [N]    f32
    float* out     = (float*)d_out;             // scalar f32
    float* partial = (float*)d_ws;

    const int n = in_sizes[1];  // N rows

    constexpr int BLOCK = 512;  // 16 wave32 waves per block
    int grid = 1024;            // 16,384 waves device-wide; 4 KiB of partials
    if (ws_size < (size_t)grid * sizeof(float)) {
        grid = (int)(ws_size / sizeof(float));
        if (grid < 1) grid = 1;
    }

    focal_partial_kernel<BLOCK><<<grid, BLOCK, 0, stream>>>(pred, gold, partial, n);
    focal_final_kernel<<<1, 256, 0, stream>>>(partial, out, grid);
}